// Attention_82317343195784
// MI455X (gfx1250) — compile-verified
//
#include <hip/hip_runtime.h>
#include <math.h>

// ---------------- problem constants (Qwen3-4B-like decode) ----------------
#define NB    32      // batch / total_q
#define NH    32      // attention heads
#define NKVH  8       // kv heads
#define NG    4       // GQA group size
#define HD    128     // head dim
#define HIDSZ 2560    // hidden size
#define MAXP  128     // pages per request
#define PS    16      // page size (tokens)

typedef __attribute__((ext_vector_type(2))) float v2f;
typedef __attribute__((ext_vector_type(8))) float v8f;

static __device__ __forceinline__ v8f v8f_zero() {
  v8f z = {0.f, 0.f, 0.f, 0.f, 0.f, 0.f, 0.f, 0.f};
  return z;
}

// D = A(16x4) * B(4x16) + C  -- fp32 WMMA
static __device__ __forceinline__ v8f wmma_f32(v2f a, v2f b, v8f c) {
  return __builtin_amdgcn_wmma_f32_16x16x4_f32(false, a, false, b, (short)0, c,
                                               false, false);
}

// =====================================================================
// Generic GEMM: C[M,N] = A[M,K] @ W[N,K]^T   (weights stored row-major (N,K))
// One wave (32 thr) per 16x16 output tile. grid = (N/16, M/16).
// Fragment layouts per CDNA5 ISA 7.12.2 (32-bit matrices):
//   A frag: lane L, reg i -> A[m0 + L%16][k + i + (L/16)*2]
//   B frag: lane L, reg i -> Bmat[k + i + (L/16)*2][n0 + L%16] = W[n0+L%16][..]
//   D frag: lane L, reg r -> C[m0 + r + (L/16)*8][n0 + L%16]
// =====================================================================
__global__ __launch_bounds__(32) void gemm_xwt(const float* __restrict__ A,
                                               const float* __restrict__ W,
                                               float* __restrict__ C,
                                               int M, int N, int K) {
  const int lane = threadIdx.x & 31;
  const int half = lane >> 4;
  const int l16  = lane & 15;
  const int n0 = blockIdx.x * 16;
  const int m0 = blockIdx.y * 16;

  const float* arow = A + (size_t)(m0 + l16) * K + half * 2;
  const float* wrow = W + (size_t)(n0 + l16) * K + half * 2;

  v8f acc = v8f_zero();
#pragma unroll 4
  for (int k = 0; k < K; k += 4) {
    float2 av = *(const float2*)(arow + k);
    float2 wv = *(const float2*)(wrow + k);
    v2f a; a.x = av.x; a.y = av.y;
    v2f b; b.x = wv.x; b.y = wv.y;
    acc = wmma_f32(a, b, acc);
  }
#pragma unroll
  for (int r = 0; r < 8; ++r)
    C[(size_t)(m0 + r + half * 8) * N + n0 + l16] = acc[r];
}

// =====================================================================
// Per-head RMSNorm + RoPE. One wave per (b, head); lane owns 4 dims.
// idx < NH            : q head  -> norm(q_w) + rope -> qout
// idx < NH+NKVH       : k head  -> norm(k_w) + rope -> knew
// else                : v head  -> plain copy       -> vnew
// rotate-half partner d^64 lives in lane^16 (d = lane*4 + i).
// =====================================================================
__global__ __launch_bounds__(32) void norm_rope(
    const float* __restrict__ qraw, const float* __restrict__ kraw,
    const float* __restrict__ vraw, const float* __restrict__ cosb,
    const float* __restrict__ sinb, const float* __restrict__ qw,
    const float* __restrict__ kw, float* __restrict__ qout,
    float* __restrict__ knew, float* __restrict__ vnew) {
  const int bidx = blockIdx.x;
  const int b   = bidx / (NH + 2 * NKVH);
  const int idx = bidx % (NH + 2 * NKVH);
  const int lane = threadIdx.x & 31;

  const float* src;
  float* dst;
  const float* w;
  if (idx < NH) {
    src = qraw + ((size_t)b * NH + idx) * HD;
    dst = qout + ((size_t)b * NH + idx) * HD;
    w = qw;
  } else if (idx < NH + NKVH) {
    const int kh = idx - NH;
    src = kraw + ((size_t)b * NKVH + kh) * HD;
    dst = knew + ((size_t)b * NKVH + kh) * HD;
    w = kw;
  } else {
    const int vh = idx - NH - NKVH;
    src = vraw + ((size_t)b * NKVH + vh) * HD;
    dst = vnew + ((size_t)b * NKVH + vh) * HD;
    w = nullptr;
  }

  float x[4];
#pragma unroll
  for (int i = 0; i < 4; ++i) x[i] = src[lane * 4 + i];

  if (w != nullptr) {
    float ss = 0.f;
#pragma unroll
    for (int i = 0; i < 4; ++i) ss += x[i] * x[i];
#pragma unroll
    for (int off = 16; off >= 1; off >>= 1) ss += __shfl_xor(ss, off, 32);
    const float rn = rsqrtf(ss * (1.0f / HD) + 1e-6f);
#pragma unroll
    for (int i = 0; i < 4; ++i) {
      const int d = lane * 4 + i;
      const float xn = x[i] * rn * w[d];
      const float part = __shfl_xor(xn, 16, 32);  // normalized x[d ^ 64]
      const float cv = cosb[b * HD + d];
      const float sv = sinb[b * HD + d];
      x[i] = xn * cv + ((lane < 16) ? -part : part) * sv;
    }
  }
#pragma unroll
  for (int i = 0; i < 4; ++i) dst[lane * 4 + i] = x[i];
}

// =====================================================================
// Paged GQA decode attention (flash-decode). One WG per (b, kv-head),
// 8 waves; wave w owns chunks w, w+8, ... (1 chunk = 1 page = 16 keys).
// S^T tile (16 keys x 16 heads, 4 live) via 32 f32 WMMAs over D=128;
// online softmax per head column with shfl_xor(16) cross-half reduce;
// P staged via per-wave LDS for PV WMMAs; final LSE merge across waves.
// New token (pos = len-1) is read from knew/vnew, never from the pools.
// =====================================================================
__global__ __launch_bounds__(256) void attn_decode(
    const float* __restrict__ qbuf, const float* __restrict__ knew,
    const float* __restrict__ vnew, const float* __restrict__ Kpool,
    const float* __restrict__ Vpool, const int* __restrict__ pages,
    const int* __restrict__ lengths, float* __restrict__ att_out) {
  const int b    = blockIdx.x / NKVH;
  const int kvh  = blockIdx.x % NKVH;
  const int tid  = threadIdx.x;
  const int wave = tid >> 5;
  const int lane = tid & 31;
  const int half = lane >> 4;
  const int l16  = lane & 15;

  __shared__ float q_smem[16][HD];        // 8 KB, heads >= NG zeroed
  __shared__ float p_smem[8][16][16];     // 8 KB, per-wave P staging
  __shared__ float o_smem[8][NG][HD];     // 16 KB, per-wave partial O
  __shared__ float m_smem[8][16];
  __shared__ float l_smem[8][16];

  for (int i = tid; i < 16 * HD; i += 256) {
    const int g = i >> 7, d = i & 127;
    q_smem[g][d] =
        (g < NG) ? qbuf[((size_t)b * NH + kvh * NG + g) * HD + d] : 0.0f;
  }
  __syncthreads();

  const int len = lengths[b];
  const int nchunks = (len + PS - 1) / PS;
  const float* knew_p = knew + ((size_t)b * NKVH + kvh) * HD;
  const float* vnew_p = vnew + ((size_t)b * NKVH + kvh) * HD;
  const size_t slot_stride = (size_t)NKVH * HD;

  float m_run = -__builtin_inff();
  float l_run = 0.0f;
  v8f o_acc[8];
#pragma unroll
  for (int t = 0; t < 8; ++t) o_acc[t] = v8f_zero();

  for (int c = wave; c < nchunks; c += 8) {
    const int page = pages[b * MAXP + c];
    const size_t pagebase =
        (size_t)page * PS * slot_stride + (size_t)kvh * HD;

    // ---- S^T = K_chunk(16xD) * Q^T(Dx16) ----
    const int kpos_a = c * PS + l16;
    const float* krow = (kpos_a == len - 1)
                            ? knew_p
                            : (Kpool + pagebase + (size_t)l16 * slot_stride);
    v8f s = v8f_zero();
#pragma unroll
    for (int ks = 0; ks < 32; ++ks) {
      const int d0 = ks * 4 + half * 2;
      float2 kv = *(const float2*)(krow + d0);
      v2f a;  a.x = kv.x;             a.y = kv.y;
      v2f bq; bq.x = q_smem[l16][d0]; bq.y = q_smem[l16][d0 + 1];
      s = wmma_f32(a, bq, s);
    }

    // ---- online softmax (score at [key = r + half*8][head = l16]) ----
    const float scale = 0.08838834764831845f;  // 1/sqrt(128)
    float vals[8];
#pragma unroll
    for (int r = 0; r < 8; ++r) {
      const int kp = c * PS + r + half * 8;
      vals[r] = (kp < len) ? s[r] * scale : -__builtin_inff();
    }
    float mc = vals[0];
#pragma unroll
    for (int r = 1; r < 8; ++r) mc = fmaxf(mc, vals[r]);
    mc = fmaxf(mc, __shfl_xor(mc, 16, 32));
    const float m_new = fmaxf(m_run, mc);
    const float corr = __expf(m_run - m_new);  // 0 on first chunk

    float p[8];
    float psum = 0.0f;
#pragma unroll
    for (int r = 0; r < 8; ++r) {
      p[r] = __expf(vals[r] - m_new);
      psum += p[r];
    }
    psum += __shfl_xor(psum, 16, 32);
    l_run = l_run * corr + psum;
    m_run = m_new;
#pragma unroll
    for (int t = 0; t < 8; ++t)
#pragma unroll
      for (int r = 0; r < 8; ++r) o_acc[t][r] *= corr;

    // stage P into per-wave LDS region: p_smem[wave][key][head]
#pragma unroll
    for (int r = 0; r < 8; ++r) p_smem[wave][r + half * 8][l16] = p[r];

    // ---- O += P(16 heads x 16 keys) * V(16 keys x D) ----
#pragma unroll
    for (int ks = 0; ks < 4; ++ks) {
      const int k0 = ks * 4 + half * 2;
      v2f a;
      a.x = p_smem[wave][k0][l16];
      a.y = p_smem[wave][k0 + 1][l16];
      const float* vr0 = (c * PS + k0 == len - 1)
                             ? vnew_p
                             : (Vpool + pagebase + (size_t)k0 * slot_stride);
      const float* vr1 =
          (c * PS + k0 + 1 == len - 1)
              ? vnew_p
              : (Vpool + pagebase + (size_t)(k0 + 1) * slot_stride);
#pragma unroll
      for (int t = 0; t < 8; ++t) {
        v2f bv;
        bv.x = vr0[t * 16 + l16];
        bv.y = vr1[t * 16 + l16];
        o_acc[t] = wmma_f32(a, bv, o_acc[t]);
      }
    }
  }

  // ---- publish per-wave partials (valid heads live in half==0, r<NG) ----
  if (half == 0) {
    m_smem[wave][l16] = m_run;
    l_smem[wave][l16] = l_run;
#pragma unroll
    for (int t = 0; t < 8; ++t)
#pragma unroll
      for (int r = 0; r < NG; ++r)
        o_smem[wave][r][t * 16 + l16] = o_acc[t][r];
  }
  __syncthreads();

  // ---- merge 8 wave partials (log-sum-exp) ----
  for (int idx = tid; idx < NG * HD; idx += 256) {
    const int g = idx >> 7, d = idx & 127;
    float m = m_smem[0][g];
#pragma unroll
    for (int w = 1; w < 8; ++w) m = fmaxf(m, m_smem[w][g]);
    float num = 0.f, den = 0.f;
#pragma unroll
    for (int w = 0; w < 8; ++w) {
      const float e = __expf(m_smem[w][g] - m);
      num += e * o_smem[w][g][d];
      den += e * l_smem[w][g];
    }
    att_out[(size_t)b * (NH * HD) + (kvh * NG + g) * HD + d] = num / den;
  }
}

// =====================================================================
// Host-side launch
// =====================================================================
extern "C" void kernel_launch(void* const* d_in, const int* in_sizes, int n_in,
                              void* d_out, int out_size, void* d_ws,
                              size_t ws_size, hipStream_t stream) {
  (void)in_sizes; (void)n_in; (void)out_size; (void)ws_size;
  const float* x     = (const float*)d_in[0];
  const float* cosb  = (const float*)d_in[1];
  const float* sinb  = (const float*)d_in[2];
  const float* Wq    = (const float*)d_in[3];
  const float* Wk    = (const float*)d_in[4];
  const float* Wv    = (const float*)d_in[5];
  const float* Wo    = (const float*)d_in[6];
  const float* qw    = (const float*)d_in[7];
  const float* kw    = (const float*)d_in[8];
  const float* Kpool = (const float*)d_in[9];
  const float* Vpool = (const float*)d_in[10];
  const int* pages   = (const int*)d_in[11];
  const int* lengths = (const int*)d_in[12];
  // d_in[13] slot_mapping: not needed (new token handled via knew/vnew)

  float* wsf   = (float*)d_ws;
  float* qraw  = wsf;                       // 32*4096
  float* kraw  = qraw + NB * NH * HD;       // 32*1024
  float* vraw  = kraw + NB * NKVH * HD;     // 32*1024
  float* qbuf  = vraw + NB * NKVH * HD;     // 32*4096
  float* knew  = qbuf + NB * NH * HD;       // 32*1024
  float* vnew  = knew + NB * NKVH * HD;     // 32*1024
  float* attout = vnew + NB * NKVH * HD;    // 32*4096

  // QKV projections (x: (32, 2560); weights (out, 2560))
  gemm_xwt<<<dim3((NH * HD) / 16, NB / 16), 32, 0, stream>>>(
      x, Wq, qraw, NB, NH * HD, HIDSZ);
  gemm_xwt<<<dim3((NKVH * HD) / 16, NB / 16), 32, 0, stream>>>(
      x, Wk, kraw, NB, NKVH * HD, HIDSZ);
  gemm_xwt<<<dim3((NKVH * HD) / 16, NB / 16), 32, 0, stream>>>(
      x, Wv, vraw, NB, NKVH * HD, HIDSZ);

  // QK-norm + RoPE, produce q_buf / k_new / v_new
  norm_rope<<<NB * (NH + 2 * NKVH), 32, 0, stream>>>(
      qraw, kraw, vraw, cosb, sinb, qw, kw, qbuf, knew, vnew);

  // paged flash-decode attention
  attn_decode<<<NB * NKVH, 256, 0, stream>>>(qbuf, knew, vnew, Kpool, Vpool,
                                             pages, lengths, attout);

  // o_proj: y(32, 2560) = attout(32, 4096) @ Wo(2560, 4096)^T
  gemm_xwt<<<dim3(HIDSZ / 16, NB / 16), 32, 0, stream>>>(
      attout, Wo, (float*)d_out, NB, HIDSZ, NH * HD);
}